// CRF_73194832658535
// MI455X (gfx1250) — compile-verified
//
#include <hip/hip_runtime.h>

// Problem constants (from reference): B=512, T=1024, K=48, START=45, STOP=46, PAD=47
#define BSZ 512
#define TLEN 1024
#define KTAG 48
#define STOP_TAG 46
#define NEGV -10000.0f
#define LDS_STRIDE 20   // padded row stride (dwords) -> conflict-free half-wave access
#define LOG2E 1.44269504088896340736f
#define LN2   0.69314718055994530942f

typedef __attribute__((ext_vector_type(2))) float v2f;
typedef __attribute__((ext_vector_type(4))) float v4f;
typedef __attribute__((ext_vector_type(8))) float v8f;

// Kernel 1: Texp[i][j] = exp(trans[i][j]).  exp(-10000) underflows to 0.0f,
// which is exactly the lse-limit behavior for the masked transition entries.
__global__ void crf_texp_kernel(const float* __restrict__ trans,
                                float* __restrict__ texp) {
    int i = blockIdx.x * blockDim.x + threadIdx.x;
    if (i < KTAG * KTAG) texp[i] = __expf(trans[i]);
}

// Kernel 2: the scan, kept internally in the log2 domain (v_exp/v_log are base-2).
// One workgroup per 16-batch tile, 3 waves per WG; wave w owns output tags
// [16w, 16w+16) and runs 12 x V_WMMA_F32_16X16X4_F32 per step (2 acc chains).
__global__ __launch_bounds__(96) void crf_scan_kernel(
    const float* __restrict__ h_tag,   // (B, T, K) f32, natural-log emissions
    const float* __restrict__ mask,    // (B, T)   f32 (exactly 0.0 or 1.0)
    const float* __restrict__ trans,   // (K, K)   f32, natural-log transitions
    const float* __restrict__ texp,    // (K, K)   f32 = exp(trans), linear domain
    float* __restrict__ out) {         // (B,)     f32, natural log

    __shared__ float a_lds[KTAG * LDS_STRIDE]; // rescaled 2^score tile, tags x 16 batch
    __shared__ float red_lds[3 * 16];          // per-wave partial row maxima

    const int lane = threadIdx.x & 31;
    const int w    = threadIdx.x >> 5;  // 0..2 : tag fragment index (M tile)
    const int lb   = lane & 15;         // batch column within tile (N)
    const int hi   = lane >> 4;         // half-wave selector
    const int b    = blockIdx.x * 16 + lb;

    // Loop-invariant A fragments: A[k] holds Texp[16w+lb][4k + 2*hi + {0,1}]
    const int arow = 16 * w + lb;
    v2f A[12];
#pragma unroll
    for (int k = 0; k < 12; ++k)
        A[k] = *(const v2f*)(texp + arow * KTAG + 4 * k + 2 * hi);

    // Score fragment in log2 units: S2[r] = score[16w + r + 8hi][lb] * log2(e)
    float S2[8];
#pragma unroll
    for (int r = 0; r < 8; ++r) {
        int tag = 16 * w + r + 8 * hi;
        S2[r] = (tag == STOP_TAG) ? 0.0f : NEGV * LOG2E;
    }

    const float* emit_base = h_tag + ((size_t)b * TLEN) * KTAG + 16 * w + 8 * hi;
    const float* mask_base = mask + (size_t)b * TLEN;

    // software-pipelined emission/mask registers (current step)
    v4f   e0   = *(const v4f*)(emit_base);
    v4f   e1   = *(const v4f*)(emit_base + 4);
    float mval = mask_base[0];

    for (int t = 0; t < TLEN; ++t) {
        // ---- issue next step's global loads early (clamped, branch-free) ----
        const int tn = (t + 1 < TLEN) ? (t + 1) : (TLEN - 1);
        const float* epn = emit_base + (size_t)tn * KTAG;
        v4f   ne0 = *(const v4f*)(epn);
        v4f   ne1 = *(const v4f*)(epn + 4);
        float nmv = mask_base[tn];
        if (t + 12 < TLEN)
            __builtin_prefetch(emit_base + (size_t)(t + 12) * KTAG, 0, 1);

        // ---- 1. per-batch row max over all 48 tags (log2 units) ----
        float mx = S2[0];
#pragma unroll
        for (int r = 1; r < 8; ++r) mx = fmaxf(mx, S2[r]);
        mx = fmaxf(mx, __shfl_xor(mx, 16, 32));   // fold complementary tag halves
        red_lds[w * 16 + lb] = mx;                // lanes l / l^16 write same value
        __syncthreads();
        float m2 = fmaxf(fmaxf(red_lds[lb], red_lds[16 + lb]), red_lds[32 + lb]);

        // ---- 2. a = 2^(score2 - m2) -> LDS (raw v_exp_f32) ----
#pragma unroll
        for (int r = 0; r < 8; ++r)
            a_lds[(16 * w + r + 8 * hi) * LDS_STRIDE + lb] =
                __builtin_amdgcn_exp2f(S2[r] - m2);
        __syncthreads();

        // ---- 3. preload all B fragments, then WMMA (2 accumulator chains) ----
        v2f Bfr[12];
#pragma unroll
        for (int k = 0; k < 12; ++k) {
            Bfr[k].x = a_lds[(4 * k + 2 * hi) * LDS_STRIDE + lb];
            Bfr[k].y = a_lds[(4 * k + 2 * hi + 1) * LDS_STRIDE + lb];
        }
        v8f V0 = {}, V1 = {};
#pragma unroll
        for (int k = 0; k < 12; k += 2) {
            V0 = __builtin_amdgcn_wmma_f32_16x16x4_f32(
                     false, A[k], false, Bfr[k], (short)0, V0, false, false);
            V1 = __builtin_amdgcn_wmma_f32_16x16x4_f32(
                     false, A[k + 1], false, Bfr[k + 1], (short)0, V1, false, false);
        }

        // ---- 4. new score2 = emit*log2e + m2 + log2(V); branch-free masked
        //         blend: S2 = fma(mask, ns - S2, S2)  (mask is exactly 0/1, so
        //         no exec divergence and EXEC stays all-ones around WMMA) ----
#pragma unroll
        for (int r = 0; r < 4; ++r) {
            float v0 = V0[r] + V1[r];
            float v1 = V0[r + 4] + V1[r + 4];
            float ns0 = fmaf(e0[r], LOG2E, m2) +
                        __builtin_amdgcn_logf(fmaxf(v0, 1e-30f));
            float ns1 = fmaf(e1[r], LOG2E, m2) +
                        __builtin_amdgcn_logf(fmaxf(v1, 1e-30f));
            S2[r]     = fmaf(mval, ns0 - S2[r],     S2[r]);
            S2[r + 4] = fmaf(mval, ns1 - S2[r + 4], S2[r + 4]);
        }
        e0 = ne0; e1 = ne1; mval = nmv;
    }

    // ---- finalize: out[b] = ln2 * lse2_i( score2[i,b] + trans[STOP][i]*log2e ) ----
    {
        const float* tp = trans + STOP_TAG * KTAG + 16 * w + 8 * hi;  // 32B-aligned
        v4f t0 = *(const v4f*)(tp);
        v4f t1 = *(const v4f*)(tp + 4);
        float f2[8];
#pragma unroll
        for (int r = 0; r < 4; ++r) {
            f2[r]     = fmaf(t0[r], LOG2E, S2[r]);
            f2[r + 4] = fmaf(t1[r], LOG2E, S2[r + 4]);
        }
        float mx = f2[0];
#pragma unroll
        for (int r = 1; r < 8; ++r) mx = fmaxf(mx, f2[r]);
        mx = fmaxf(mx, __shfl_xor(mx, 16, 32));
        red_lds[w * 16 + lb] = mx;
        __syncthreads();
        float M2 = fmaxf(fmaxf(red_lds[lb], red_lds[16 + lb]), red_lds[32 + lb]);

        float s = 0.0f;
#pragma unroll
        for (int r = 0; r < 8; ++r) s += __builtin_amdgcn_exp2f(f2[r] - M2);
        s += __shfl_xor(s, 16, 32);
        a_lds[w * 16 + lb] = s;   // reuse a_lds for partial sums
        __syncthreads();
        if (w == 0 && lane < 16)
            out[b] = (M2 + __builtin_amdgcn_logf(a_lds[lb] + a_lds[16 + lb] +
                                                 a_lds[32 + lb])) * LN2;
    }
}

extern "C" void kernel_launch(void* const* d_in, const int* in_sizes, int n_in,
                              void* d_out, int out_size, void* d_ws, size_t ws_size,
                              hipStream_t stream) {
    const float* h_tag = (const float*)d_in[0];
    const float* mask  = (const float*)d_in[1];
    const float* trans = (const float*)d_in[2];
    float* out  = (float*)d_out;
    float* texp = (float*)d_ws;  // 48*48 floats = 9216 B scratch

    crf_texp_kernel<<<(KTAG * KTAG + 255) / 256, 256, 0, stream>>>(trans, texp);
    crf_scan_kernel<<<BSZ / 16, 96, 0, stream>>>(h_tag, mask, trans, texp, out);
}